// SqueezeAndExcitation_9955734192437
// MI455X (gfx1250) — compile-verified
//
#include <hip/hip_runtime.h>
#include <math.h>

typedef float v2f __attribute__((ext_vector_type(2)));
typedef float v8f __attribute__((ext_vector_type(8)));

#define BB   64
#define CC   384
#define CHID 96
#define HW   3136   // 56*56
#define HW4  784    // HW/4
#define LDH  100    // padded LDS row stride for hidden buffer

// ---------------- Kernel 1: global average pool over H*W per (b,c) ----------------
__global__ __launch_bounds__(256) void se_pool_kernel(const float* __restrict__ x,
                                                      float* __restrict__ se) {
  __shared__ float sdata[256];
  const int plane = blockIdx.x;  // b*CC + c
  const float4* xp = (const float4*)(x + (size_t)plane * HW);
  float s = 0.f;
  for (int i = threadIdx.x; i < HW4; i += 256) {
    float4 v = xp[i];
    s += v.x + v.y + v.z + v.w;
  }
  sdata[threadIdx.x] = s;
  __syncthreads();
  for (int off = 128; off > 0; off >>= 1) {
    if ((int)threadIdx.x < off) sdata[threadIdx.x] += sdata[threadIdx.x + off];
    __syncthreads();
  }
  if (threadIdx.x == 0) se[plane] = sdata[0] * (1.0f / (float)HW);
}

// ---------------- Kernel 2: fused 2-layer MLP + sigmoid via f32 WMMA ----------------
// One workgroup, 24 waves. Layer1: 24 tiles (one per wave). Layer2: 96 tiles (4 per wave).
__global__ __launch_bounds__(768) void se_mlp_kernel(const float* __restrict__ se,
                                                     const float* __restrict__ w_reduce,
                                                     const float* __restrict__ b_reduce,
                                                     const float* __restrict__ w_expand,
                                                     const float* __restrict__ b_expand,
                                                     float* __restrict__ gate) {
  __shared__ float hbuf[64 * LDH];
  const int lane = threadIdx.x & 31;
  const int wave = threadIdx.x >> 5;   // 0..23
  const int half = lane >> 4;          // 0 or 1
  const int l16  = lane & 15;

  // ---- Layer 1: h = relu(se[64,384] @ w_reduce[96,384]^T + b_reduce) ----
  // Tiles: 4 (M) x 6 (N); wave -> (mt, nt)
  {
    const int mt = wave / 6, nt = wave % 6;
    const int m0 = mt * 16, n0 = nt * 16;
    v8f acc = {};
    // A frag: element [m = m0+l16, k = k0 + half*2 + v]
    const float* arow = se + (size_t)(m0 + l16) * CC + half * 2;
    // B frag: B[k,n] = w_reduce[n*CC + k], n = n0+l16
    const float* brow = w_reduce + (size_t)(n0 + l16) * CC + half * 2;
    for (int k0 = 0; k0 < CC; k0 += 4) {
      v2f a = *(const v2f*)(arow + k0);
      v2f b = *(const v2f*)(brow + k0);
      acc = __builtin_amdgcn_wmma_f32_16x16x4_f32(false, a, false, b,
                                                  (short)0, acc, false, false);
    }
    const float bias = b_reduce[n0 + l16];
#pragma unroll
    for (int r = 0; r < 8; ++r) {
      float v = acc[r] + bias;
      // D element: [m = m0 + r + half*8, n = n0 + l16]
      hbuf[(m0 + r + half * 8) * LDH + n0 + l16] = fmaxf(v, 0.f);
    }
  }
  __syncthreads();

  // ---- Layer 2: gate = sigmoid(h[64,96] @ w_expand[384,96]^T + b_expand) ----
  // Tiles: 4 (M) x 24 (N); wave owns column strip nt = wave, mt = 0..3
  {
    const int n0 = wave * 16;
    const float bias = b_expand[n0 + l16];
    const float* brow = w_expand + (size_t)(n0 + l16) * CHID + half * 2;
#pragma unroll
    for (int mt = 0; mt < 4; ++mt) {
      const int m0 = mt * 16;
      v8f acc = {};
      const float* arow = hbuf + (m0 + l16) * LDH + half * 2;
      for (int k0 = 0; k0 < CHID; k0 += 4) {
        v2f a = *(const v2f*)(arow + k0);
        v2f b = *(const v2f*)(brow + k0);
        acc = __builtin_amdgcn_wmma_f32_16x16x4_f32(false, a, false, b,
                                                    (short)0, acc, false, false);
      }
#pragma unroll
      for (int r = 0; r < 8; ++r) {
        float v = acc[r] + bias;
        gate[(size_t)(m0 + r + half * 8) * CC + n0 + l16] = 1.0f / (1.0f + expf(-v));
      }
    }
  }
}

// ---------------- Kernel 3: out = sigmoid-gate broadcast multiply ----------------
__global__ __launch_bounds__(256) void se_gate_kernel(const float* __restrict__ x,
                                                      const float* __restrict__ gate,
                                                      float* __restrict__ out) {
  const size_t idx = (size_t)blockIdx.x * 256 + threadIdx.x;  // float4 index
  const int plane = (int)(idx / HW4);                         // b*CC + c
  const float g = gate[plane];
  float4 v = ((const float4*)x)[idx];
  v.x *= g; v.y *= g; v.z *= g; v.w *= g;
  ((float4*)out)[idx] = v;
}

extern "C" void kernel_launch(void* const* d_in, const int* in_sizes, int n_in,
                              void* d_out, int out_size, void* d_ws, size_t ws_size,
                              hipStream_t stream) {
  const float* x        = (const float*)d_in[0];
  const float* w_reduce = (const float*)d_in[1];
  const float* b_reduce = (const float*)d_in[2];
  const float* w_expand = (const float*)d_in[3];
  const float* b_expand = (const float*)d_in[4];
  float* out  = (float*)d_out;
  float* se   = (float*)d_ws;            // [64*384]
  float* gate = se + BB * CC;            // [64*384]

  se_pool_kernel<<<BB * CC, 256, 0, stream>>>(x, se);
  se_mlp_kernel<<<1, 768, 0, stream>>>(se, w_reduce, b_reduce, w_expand, b_expand, gate);
  se_gate_kernel<<<(BB * CC * HW4) / 256, 256, 0, stream>>>(x, gate, out);
}